// GCNNet1_41016937677161
// MI455X (gfx1250) — compile-verified
//
#include <hip/hip_runtime.h>
#include <hip/hip_bf16.h>

// ---------------- constants (match reference) ----------------
#define N_NODES 50000
#define N_EDGES 800000
#define N_GRAPHS 64
#define IN_DIM 128
#define HID 146
#define HP 160            // HID padded to multiple of 16
#define NT (HP / 16)      // 10 column tiles

typedef __attribute__((ext_vector_type(2))) float v2f;
typedef __attribute__((ext_vector_type(8))) float v8f;

// ---------------------------------------------------------------
// Pad a [srcRows x srcCols] weight into [gridDim.x x HP], zero-filled.
// blockDim.x == HP, gridDim.x == padded K rows.
__global__ void pad_weight(const float* __restrict__ W, float* __restrict__ Wp,
                           int srcRows, int srcCols) {
    int k = blockIdx.x;
    int c = threadIdx.x;
    float v = (k < srcRows && c < srcCols) ? W[(size_t)k * srcCols + c] : 0.0f;
    Wp[(size_t)k * HP + c] = v;
}

// ---------------------------------------------------------------
// C[M x HP] = A[M x lda] @ B[K x HP]  (+bias on cols < HID if bias != null)
// One wave per 16-row strip; 10 accumulators cover all HP columns.
// Uses V_WMMA_F32_16X16X4_F32 (f32-exact matrix core path, wave32).
__global__ void __launch_bounds__(256)
gemm_wmma_f32(const float* __restrict__ A, int lda, int K,
              const float* __restrict__ B, float* __restrict__ C,
              const float* __restrict__ bias, int M) {
    int wave = (blockIdx.x * blockDim.x + threadIdx.x) >> 5;
    int lane = threadIdx.x & 31;
    int m0   = wave * 16;
    if (m0 >= M) return;            // wave-uniform; active waves keep EXEC all-1s

    int half = lane >> 4;           // 0 or 1
    int l15  = lane & 15;
    int m    = m0 + l15;            // A-matrix row for this lane
    int kk   = half * 2;            // lane's K sub-pair within the 4-wide step

    v8f acc[NT];
    #pragma unroll
    for (int t = 0; t < NT; ++t)
        #pragma unroll
        for (int r = 0; r < 8; ++r) acc[t][r] = 0.0f;

    for (int k = 0; k < K; k += 4) {
        // A fragment 16x4: lane -> {A[m][k+kk], A[m][k+kk+1]}
        const float* ap = A + (size_t)m * lda + k + kk;
        v2f a;
        a.x = ap[0];
        a.y = ap[1];
        // B fragments 4x16 per tile: VGPR0 rows {kk}, VGPR1 rows {kk+1}
        const float* b0 = B + (size_t)(k + kk) * HP + l15;
        const float* b1 = b0 + HP;
        #pragma unroll
        for (int t = 0; t < NT; ++t) {
            v2f b;
            b.x = b0[t * 16];
            b.y = b1[t * 16];
            acc[t] = __builtin_amdgcn_wmma_f32_16x16x4_f32(
                false, a, false, b, (short)0, acc[t], false, false);
        }
    }

    // D layout: VGPR r -> row (m0 + r + 8*half), col = t*16 + l15
    int rowBase = m0 + half * 8;
    #pragma unroll
    for (int t = 0; t < NT; ++t) {
        int col = t * 16 + l15;
        float bb = 0.0f;
        if (bias != nullptr && col < HID) bb = bias[col];
        #pragma unroll
        for (int r = 0; r < 8; ++r)
            C[(size_t)(rowBase + r) * HP + col] = acc[t][r] + bb;
    }
}

// ---------------------------------------------------------------
// H[n][c] = bias[c] for c < HID else 0.   gridDim.x == N_NODES, blockDim.x == HP
__global__ void init_bias_rows(float* __restrict__ H, const float* __restrict__ b) {
    int n = blockIdx.x;
    int c = threadIdx.x;
    H[(size_t)n * HP + c] = (c < HID) ? b[c] : 0.0f;
}

// ---------------------------------------------------------------
// One wave per edge: H[dst[e]] += T[src[e]] (148 channels; pads are exact zero).
__global__ void __launch_bounds__(256)
scatter_edges(const float* __restrict__ T, const int* __restrict__ src,
              const int* __restrict__ dst, float* __restrict__ H, int nE) {
    int gid  = blockIdx.x * blockDim.x + threadIdx.x;
    int e    = gid >> 5;
    int lane = threadIdx.x & 31;
    if (e >= nE) return;
    int s = src[e];
    int d = dst[e];
    const float* trow = T + (size_t)s * HP;
    float* hrow = H + (size_t)d * HP;
    for (int c = lane * 4; c < 148; c += 128) {
        float4 v = *(const float4*)(trow + c);
        unsafeAtomicAdd(hrow + c + 0, v.x);
        unsafeAtomicAdd(hrow + c + 1, v.y);
        unsafeAtomicAdd(hrow + c + 2, v.z);
        unsafeAtomicAdd(hrow + c + 3, v.w);
    }
}

// ---------------------------------------------------------------
__global__ void relu_inplace(float* __restrict__ H, int n) {
    int i = blockIdx.x * blockDim.x + threadIdx.x;
    if (i < n) {
        float v = H[i];
        H[i] = v > 0.0f ? v : 0.0f;
    }
}

// ---------------------------------------------------------------
__global__ void pool_zero(float* __restrict__ sums, float* __restrict__ cnts) {
    int i = blockIdx.x * blockDim.x + threadIdx.x;
    if (i < N_GRAPHS * HP) sums[i] = 0.0f;
    if (i < N_GRAPHS) cnts[i] = 0.0f;
}

// One wave per node: sums[batch[n]] += H[n]; cnts[batch[n]] += 1.
__global__ void __launch_bounds__(256)
pool_scatter(const float* __restrict__ H, const int* __restrict__ batch,
             float* __restrict__ sums, float* __restrict__ cnts, int M) {
    int gid  = blockIdx.x * blockDim.x + threadIdx.x;
    int n    = gid >> 5;
    int lane = threadIdx.x & 31;
    if (n >= M) return;
    int g = batch[n];
    const float* hrow = H + (size_t)n * HP;
    float* srow = sums + (size_t)g * HP;
    for (int c = lane * 4; c < 148; c += 128) {
        float4 v = *(const float4*)(hrow + c);
        unsafeAtomicAdd(srow + c + 0, v.x);
        unsafeAtomicAdd(srow + c + 1, v.y);
        unsafeAtomicAdd(srow + c + 2, v.z);
        unsafeAtomicAdd(srow + c + 3, v.w);
    }
    if (lane == 0) unsafeAtomicAdd(cnts + g, 1.0f);
}

// out[g][c] = sums[g][c] / max(cnt[g], 1).  gridDim.x == N_GRAPHS, blockDim.x == HP
__global__ void pool_out(const float* __restrict__ sums, const float* __restrict__ cnts,
                         float* __restrict__ out) {
    int g = blockIdx.x;
    int c = threadIdx.x;
    if (c < HID) {
        float cnt = cnts[g];
        cnt = cnt > 1.0f ? cnt : 1.0f;
        out[(size_t)g * HID + c] = sums[(size_t)g * HP + c] / cnt;
    }
}

// ---------------------------------------------------------------
extern "C" void kernel_launch(void* const* d_in, const int* in_sizes, int n_in,
                              void* d_out, int out_size, void* d_ws, size_t ws_size,
                              hipStream_t stream) {
    const float* x     = (const float*)d_in[0];   // [N_NODES, IN_DIM]
    const int*   ei    = (const int*)d_in[1];     // [2, N_EDGES]
    const int*   batch = (const int*)d_in[2];     // [N_NODES]
    const float* W_emb = (const float*)d_in[3];   // [IN_DIM, HID]
    const float* b_emb = (const float*)d_in[4];   // [HID]
    const float* W_gcn = (const float*)d_in[5];   // [4, HID, HID]
    const float* b_gcn = (const float*)d_in[6];   // [4, HID]
    float* out = (float*)d_out;                   // [N_GRAPHS, HID]

    const int* src = ei;
    const int* dst = ei + N_EDGES;

    // workspace layout (floats): hPad | tPad | Wpad | sums | cnts
    float* ws   = (float*)d_ws;
    float* hPad = ws;
    float* tPad = hPad + (size_t)N_NODES * HP;
    float* Wpad = tPad + (size_t)N_NODES * HP;
    float* sums = Wpad + (size_t)HP * HP;
    float* cnts = sums + (size_t)N_GRAPHS * HP;

    const int gemmWaves  = N_NODES / 16;                     // 3125
    const int gemmBlocks = (gemmWaves + 7) / 8;              // 8 waves / 256-thread block
    const int edgeBlocks = (N_EDGES * 32 + 255) / 256;       // one wave per edge
    const int nodeBlocks = (N_NODES * 32 + 255) / 256;       // one wave per node
    const int reluBlocks = (N_NODES * HP + 255) / 256;

    // Embedding: h = x @ W_emb + b_emb
    pad_weight<<<dim3(IN_DIM), dim3(HP), 0, stream>>>(W_emb, Wpad, IN_DIM, HID);
    gemm_wmma_f32<<<dim3(gemmBlocks), dim3(256), 0, stream>>>(
        x, IN_DIM, IN_DIM, Wpad, hPad, b_emb, N_NODES);

    // 4 x GCNConv: t = h @ W; h = relu(scatter_add(t[src] -> dst) + b)
    for (int i = 0; i < 4; ++i) {
        pad_weight<<<dim3(HP), dim3(HP), 0, stream>>>(
            W_gcn + (size_t)i * HID * HID, Wpad, HID, HID);
        gemm_wmma_f32<<<dim3(gemmBlocks), dim3(256), 0, stream>>>(
            hPad, HP, HP, Wpad, tPad, nullptr, N_NODES);
        init_bias_rows<<<dim3(N_NODES), dim3(HP), 0, stream>>>(
            hPad, b_gcn + (size_t)i * HID);
        scatter_edges<<<dim3(edgeBlocks), dim3(256), 0, stream>>>(
            tPad, src, dst, hPad, N_EDGES);
        relu_inplace<<<dim3(reluBlocks), dim3(256), 0, stream>>>(
            hPad, N_NODES * HP);
    }

    // Global mean pool
    pool_zero<<<dim3((N_GRAPHS * HP + 255) / 256 + 1), dim3(256), 0, stream>>>(sums, cnts);
    pool_scatter<<<dim3(nodeBlocks), dim3(256), 0, stream>>>(hPad, batch, sums, cnts, N_NODES);
    pool_out<<<dim3(N_GRAPHS), dim3(HP), 0, stream>>>(sums, cnts, out);
}